// SR_knnModel_23519240912937
// MI455X (gfx1250) — compile-verified
//
#include <hip/hip_runtime.h>
#include <cstdint>

// Problem constants (match reference)
#define B_DIM 8
#define S_DIM 128
#define V_DIM 32000
#define K_DIM 32
#define ROWS (B_DIM * S_DIM)     // 1024
#define VEC4 (V_DIM / 4)         // 8000 float4 chunks per row
#define BS 512                   // 16 wave32 waves per block

// One workgroup per (b,s) row. Row (128 KB) is staged in LDS via gfx1250
// async global->LDS copies, processed in place, then streamed back with
// async LDS->global stores.
__global__ void knn_combine_row_kernel(
    const float* __restrict__ logits,   // [ROWS, V]
    const int*   __restrict__ o_vals,   // [ROWS, K]
    const float* __restrict__ o_dists,  // [ROWS, K]
    const int*   __restrict__ c_vals,   // [ROWS, K]
    const float* __restrict__ c_dists,  // [ROWS, K]
    const int*   __restrict__ prev_w,   // [ROWS]
    const float* __restrict__ o_lam,    // [1]
    const float* __restrict__ c_lam,    // [1]
    const float* __restrict__ o_temp,   // [1]
    const float* __restrict__ c_temp,   // [1]
    float*       __restrict__ out)      // [ROWS, V]
{
    extern __shared__ float smem[];
    float* row = smem;                  // V_DIM floats (128 KB)
    float* red = smem + V_DIM;          // 16 per-wave partials
    float* bc  = red + 16;              // [0]=row max, [1]=row sum

    const int r    = blockIdx.x;
    const int tid  = threadIdx.x;
    const int wave = tid >> 5;
    const int lane = tid & 31;

    const float* grow = logits + (size_t)r * V_DIM;
    float*       orow = out    + (size_t)r * V_DIM;

    // Low 32 bits of a generic pointer into LDS == LDS byte offset
    const uint32_t lds_base = (uint32_t)(uintptr_t)row;

    // ---------------- Phase A: async copy row -> LDS (ASYNCcnt path) -----
    for (int i = tid; i < VEC4; i += BS) {
        uint32_t    la = lds_base + (uint32_t)i * 16u;
        const void* ga = (const void*)(grow + i * 4);
        asm volatile("global_load_async_to_lds_b128 %0, %1, off"
                     :: "v"(la), "v"(ga) : "memory");
    }
    asm volatile("s_wait_asynccnt 0" ::: "memory");
    __syncthreads();

    // ---------------- Phase B: row max (wave32 shuffles) ------------------
    const float4* row4 = (const float4*)row;
    float m = -INFINITY;
    for (int i = tid; i < VEC4; i += BS) {
        float4 v = row4[i];
        m = fmaxf(m, fmaxf(fmaxf(v.x, v.y), fmaxf(v.z, v.w)));
    }
    for (int off = 16; off > 0; off >>= 1)
        m = fmaxf(m, __shfl_xor(m, off, 32));
    if (lane == 0) red[wave] = m;
    __syncthreads();
    if (tid == 0) {
        float mm = red[0];
        for (int w = 1; w < BS / 32; ++w) mm = fmaxf(mm, red[w]);
        bc[0] = mm;
    }
    __syncthreads();
    const float rmax = bc[0];

    // ---------------- Phase C: row sum of exp -----------------------------
    float s = 0.0f;
    for (int i = tid; i < VEC4; i += BS) {
        float4 v = row4[i];
        s += __expf(v.x - rmax) + __expf(v.y - rmax) +
             __expf(v.z - rmax) + __expf(v.w - rmax);
    }
    for (int off = 16; off > 0; off >>= 1)
        s += __shfl_xor(s, off, 32);
    if (lane == 0) red[wave] = s;
    __syncthreads();
    if (tid == 0) {
        float ss = 0.0f;
        for (int w = 0; w < BS / 32; ++w) ss += red[w];
        bc[1] = ss;
    }
    __syncthreads();
    const float inv_sum = 1.0f / bc[1];

    // ---------------- Uniform per-row scalars -----------------------------
    const int  pw = prev_w[r];
    const bool mo = (pw <= 88) || (pw >= 91 && pw <= 291);
    const bool mc = (pw == 89) || (pw == 90) || (pw >= 292);
    const float lo = o_lam[0];
    const float lc = c_lam[0];
    const float coeff = (mo ? (1.0f - lo) : 0.0f) + (mc ? (1.0f - lc) : 0.0f);

    // ---------------- Phase D: transform row in place ---------------------
    // row[j] = coeff * softmax(logits)[j]
    float4* row4w = (float4*)row;
    const float cs = coeff * inv_sum;
    for (int i = tid; i < VEC4; i += BS) {
        float4 v = row4w[i];
        v.x = cs * __expf(v.x - rmax);
        v.y = cs * __expf(v.y - rmax);
        v.z = cs * __expf(v.z - rmax);
        v.w = cs * __expf(v.w - rmax);
        row4w[i] = v;
    }
    __syncthreads();

    // ---------------- Phase E: kNN softmax + LDS scatter ------------------
    // wave 0 -> optor table, wave 1 -> const table; K=32 fits one wave32.
    if (wave < 2) {
        const bool   active = (wave == 0) ? mo : mc;
        const int*   vals   = (wave == 0) ? o_vals  : c_vals;
        const float* dists  = (wave == 0) ? o_dists : c_dists;
        const float  temp   = (wave == 0) ? o_temp[0] : c_temp[0];
        const float  lam    = (wave == 0) ? lo : lc;

        const float x = -dists[(size_t)r * K_DIM + lane] / temp;
        float km = x;
        for (int off = 16; off > 0; off >>= 1)
            km = fmaxf(km, __shfl_xor(km, off, 32));
        const float e = __expf(x - km);
        float ks = e;
        for (int off = 16; off > 0; off >>= 1)
            ks += __shfl_xor(ks, off, 32);
        if (active) {
            const float w = lam * (e / ks);
            const int   v = vals[(size_t)r * K_DIM + lane];
            atomicAdd(&row[v], w);   // ds_add_f32 (LDS atomic)
        }
    }
    __syncthreads();

    // ---------------- Phase F: async store LDS -> global ------------------
    for (int i = tid; i < VEC4; i += BS) {
        uint32_t la = lds_base + (uint32_t)i * 16u;
        void*    ga = (void*)(orow + i * 4);
        asm volatile("global_store_async_from_lds_b128 %0, %1, off"
                     :: "v"(ga), "v"(la) : "memory");
    }
    asm volatile("s_wait_asynccnt 0" ::: "memory");
}

extern "C" void kernel_launch(void* const* d_in, const int* in_sizes, int n_in,
                              void* d_out, int out_size, void* d_ws, size_t ws_size,
                              hipStream_t stream) {
    (void)in_sizes; (void)n_in; (void)out_size; (void)d_ws; (void)ws_size;

    const float* logits  = (const float*)d_in[0];
    const int*   o_vals  = (const int*)  d_in[1];
    const float* o_dists = (const float*)d_in[2];
    const int*   c_vals  = (const int*)  d_in[3];
    const float* c_dists = (const float*)d_in[4];
    const int*   prev_w  = (const int*)  d_in[5];
    const float* o_lam   = (const float*)d_in[6];
    const float* c_lam   = (const float*)d_in[7];
    const float* o_temp  = (const float*)d_in[8];
    const float* c_temp  = (const float*)d_in[9];
    float*       out     = (float*)d_out;

    const size_t smem_bytes = (size_t)(V_DIM + 18) * sizeof(float); // ~128 KB
    knn_combine_row_kernel<<<dim3(ROWS), dim3(BS), smem_bytes, stream>>>(
        logits, o_vals, o_dists, c_vals, c_dists, prev_w,
        o_lam, c_lam, o_temp, c_temp, out);
}